// ES_Module_31662498906804
// MI455X (gfx1250) — compile-verified
//
#include <hip/hip_runtime.h>
#include <hip/hip_bf16.h>
#include <math.h>

typedef float v2f __attribute__((ext_vector_type(2)));
typedef float v8f __attribute__((ext_vector_type(8)));

#define B_  8
#define C4_ 64
#define N_  512
#define L4_ 12
#define CQ_ 16

// ---- workspace layout (floats) ----
constexpr size_t OFS_FC  = 0;                                  // [B][N][CQ][L4]
constexpr size_t OFS_FL  = OFS_FC  + (size_t)B_*N_*CQ_*L4_;    // [B][CQ][N]
constexpr size_t OFS_INV = OFS_FL  + (size_t)B_*CQ_*N_;        // [B] (pad 16)
constexpr size_t OFS_M   = OFS_INV + 16;                       // [B][N][C4][CQ]
constexpr size_t OFS_A   = OFS_M   + (size_t)B_*N_*C4_*CQ_;    // [B][N][K=N]
constexpr size_t OFS_T   = OFS_A   + (size_t)B_*N_*N_;         // [B][K][C4*CQ]

// ---------------------------------------------------------------------------
// K1: Fc[b,n,q,l] = sum_c W[q,c] x[b,c,n,l] + bc[q];  Fl[b,q,n] = Fc[..,l=11]
//     M[b,n,c,q] = sum_l x[b,c,n,l] * Fc[b,n,q,l]
// ---------------------------------------------------------------------------
__global__ __launch_bounds__(256)
void k_fc_m(const float* __restrict__ x, const float* __restrict__ Wc,
            const float* __restrict__ bc, float* __restrict__ ws) {
  __shared__ float sX[C4_*L4_];   // 768
  __shared__ float sW[CQ_*C4_];   // 1024
  __shared__ float sFc[CQ_*L4_];  // 192
  const int tid = threadIdx.x;
  const int b = blockIdx.x / N_;
  const int n = blockIdx.x % N_;
  for (int i = tid; i < CQ_*C4_; i += 256) sW[i] = Wc[i];
  for (int i = tid; i < C4_*L4_; i += 256) {
    int c = i / L4_, l = i % L4_;
    sX[i] = x[((size_t)(b*C4_ + c)*N_ + n)*L4_ + l];
  }
  __syncthreads();
  float* Fc = ws + OFS_FC + (size_t)(b*N_ + n)*(CQ_*L4_);
  float* Fl = ws + OFS_FL + (size_t)b*CQ_*N_;
  if (tid < CQ_*L4_) {
    int q = tid / L4_, l = tid % L4_;
    float f = bc[q];
    #pragma unroll 8
    for (int c = 0; c < C4_; ++c) f += sW[q*C4_ + c] * sX[c*L4_ + l];
    sFc[tid] = f;
    Fc[tid]  = f;
    if (l == L4_ - 1) Fl[q*N_ + n] = f;
  }
  __syncthreads();
  float* M = ws + OFS_M + (size_t)(b*N_ + n)*(C4_*CQ_);
  for (int i = tid; i < C4_*CQ_; i += 256) {
    int c = i / CQ_, q = i % CQ_;
    float m = 0.f;
    #pragma unroll
    for (int l = 0; l < L4_; ++l) m += sX[c*L4_ + l] * sFc[q*L4_ + l];
    M[i] = m;
  }
}

// ---------------------------------------------------------------------------
// K2: inv_l2[b] = 1 / sum_q sqrt(sum_n Fl^2) * sqrt(sum_{n,l} Fc^2)
// ---------------------------------------------------------------------------
__global__ __launch_bounds__(256)
void k_l2(float* __restrict__ ws) {
  __shared__ float pyy[CQ_*L4_];
  __shared__ float pxx[256];
  __shared__ float syy[CQ_], sxx[CQ_];
  const int tid = threadIdx.x;
  const int b = blockIdx.x;
  const float* Fc = ws + OFS_FC + (size_t)b*N_*CQ_*L4_;
  const float* Fl = ws + OFS_FL + (size_t)b*CQ_*N_;
  if (tid < CQ_*L4_) {
    float s = 0.f;
    for (int n = 0; n < N_; ++n) { float v = Fc[(size_t)n*(CQ_*L4_) + tid]; s += v*v; }
    pyy[tid] = s;
  }
  {
    int q = tid >> 4, sub = tid & 15;       // 16 threads per q, 32 n each
    float s = 0.f;
    for (int j = 0; j < 32; ++j) { float v = Fl[q*N_ + sub*32 + j]; s += v*v; }
    pxx[tid] = s;
  }
  __syncthreads();
  if (tid < CQ_) {
    float s = 0.f;
    for (int l = 0; l < L4_; ++l) s += pyy[tid*L4_ + l];
    syy[tid] = sqrtf(s);
    float s2 = 0.f;
    for (int j = 0; j < 16; ++j) s2 += pxx[tid*16 + j];
    sxx[tid] = sqrtf(s2);
  }
  __syncthreads();
  if (tid == 0) {
    float l2 = 0.f;
    for (int q = 0; q < CQ_; ++q) l2 += sxx[q]*syy[q];
    ws[OFS_INV + b] = 1.0f / l2;
  }
}

// ---------------------------------------------------------------------------
// K3: per (b,n): D = M_bn(64x16) @ G_b(16x512) via v_wmma_f32_16x16x4_f32,
//     A[b,n,k] = relu(tanh(max_c D[c,k] * inv_l2[b]))
// ---------------------------------------------------------------------------
__global__ __launch_bounds__(256)
void k_corr_max(float* __restrict__ ws) {
  __shared__ float sM[C4_*CQ_];   // 1024
  __shared__ float sG[CQ_*N_];    // 8192 (32KB)
  const int tid = threadIdx.x;
  const int b = blockIdx.x / N_;
  const int n = blockIdx.x % N_;
  const float* M = ws + OFS_M + (size_t)(b*N_ + n)*(C4_*CQ_);
  const float* G = ws + OFS_FL + (size_t)b*CQ_*N_;
  for (int i = tid; i < C4_*CQ_; i += 256) sM[i] = M[i];
  for (int i = tid; i < CQ_*N_;  i += 256) sG[i] = G[i];
  __syncthreads();
  const float inv = ws[OFS_INV + b];
  const int wave = tid >> 5, lane = tid & 31;
  const int m = lane & 15, hi = lane >> 4;
  float* Arow = ws + OFS_A + (size_t)(b*N_ + n)*N_;
  #pragma unroll
  for (int j = 0; j < 4; ++j) {          // 4 column tiles of 16 per wave
    const int col = wave*64 + j*16 + m;  // m == lane&15 doubles as N index
    float cmax = -INFINITY;
    #pragma unroll
    for (int r = 0; r < 4; ++r) {        // 4 row tiles (c blocks of 16)
      v8f acc = {0.f,0.f,0.f,0.f,0.f,0.f,0.f,0.f};
      #pragma unroll
      for (int s = 0; s < 4; ++s) {      // K = 16 -> 4 steps of 4
        v2f a, bf;
        a.x  = sM[(r*16 + m)*CQ_ + s*4 + 2*hi];
        a.y  = sM[(r*16 + m)*CQ_ + s*4 + 2*hi + 1];
        bf.x = sG[(s*4 + 2*hi    )*N_ + col];
        bf.y = sG[(s*4 + 2*hi + 1)*N_ + col];
        acc = __builtin_amdgcn_wmma_f32_16x16x4_f32(
            false, a, false, bf, (short)0, acc, false, false);
      }
      #pragma unroll
      for (int v = 0; v < 8; ++v) cmax = fmaxf(cmax, acc[v]);
    }
    // lanes L and L+16 hold the two row halves of the same column
    cmax = fmaxf(cmax, __shfl_xor(cmax, 16, 32));
    float av = tanhf(cmax * inv);
    av = av > 0.f ? av : 0.f;
    if (lane < 16) Arow[col] = av;
  }
}

// ---------------------------------------------------------------------------
// K4: per b: T(512x1024) = A_b(512x512) @ M2_b(512x1024), tiled WMMA GEMM.
//     Block: 64 rows x 128 cols; 8 waves in 2x4; wave: 2x2 16x16 tiles.
// ---------------------------------------------------------------------------
__global__ __launch_bounds__(256)
void k_t_gemm(float* __restrict__ ws) {
  __shared__ float sA[64*33];    // [row(k)][kk(n)] padded to 33
  __shared__ float sB[32*128];   // [kk(n)][col(cq)]
  const int tid = threadIdx.x;
  const int b   = blockIdx.x / 64;
  const int rem = blockIdx.x % 64;
  const int rowbase = (rem / 8) * 64;
  const int colbase = (rem % 8) * 128;
  const float* Afull = ws + OFS_A + (size_t)b*N_*N_;
  const float* Mfull = ws + OFS_M + (size_t)b*N_*(C4_*CQ_);
  float*       T     = ws + OFS_T + (size_t)b*N_*(C4_*CQ_);
  const int wave = tid >> 5, lane = tid & 31;
  const int wrow = wave >> 2, wcol = wave & 3;
  const int m = lane & 15, hi = lane >> 4;
  v8f acc[2][2];
  #pragma unroll
  for (int rt = 0; rt < 2; ++rt)
    #pragma unroll
    for (int ct = 0; ct < 2; ++ct)
      acc[rt][ct] = (v8f){0.f,0.f,0.f,0.f,0.f,0.f,0.f,0.f};

  for (int kc = 0; kc < 16; ++kc) {      // K = 512 in chunks of 32
    for (int i = tid; i < 64*32; i += 256) {
      int row = i >> 5, kk = i & 31;
      sA[row*33 + kk] = Afull[(size_t)(rowbase + row)*N_ + kc*32 + kk];
    }
    for (int i = tid; i < 32*128; i += 256) {
      int kk = i >> 7, col = i & 127;
      sB[i] = Mfull[(size_t)(kc*32 + kk)*(C4_*CQ_) + colbase + col];
    }
    __syncthreads();
    #pragma unroll
    for (int s = 0; s < 8; ++s) {        // 8 k-steps of 4
      v2f af[2], bf[2];
      #pragma unroll
      for (int rt = 0; rt < 2; ++rt) {
        int row = wrow*32 + rt*16 + m;
        af[rt].x = sA[row*33 + s*4 + 2*hi];
        af[rt].y = sA[row*33 + s*4 + 2*hi + 1];
      }
      #pragma unroll
      for (int ct = 0; ct < 2; ++ct) {
        int col = wcol*32 + ct*16 + m;
        bf[ct].x = sB[(s*4 + 2*hi    )*128 + col];
        bf[ct].y = sB[(s*4 + 2*hi + 1)*128 + col];
      }
      #pragma unroll
      for (int rt = 0; rt < 2; ++rt)
        #pragma unroll
        for (int ct = 0; ct < 2; ++ct)
          acc[rt][ct] = __builtin_amdgcn_wmma_f32_16x16x4_f32(
              false, af[rt], false, bf[ct], (short)0, acc[rt][ct], false, false);
    }
    __syncthreads();
  }
  #pragma unroll
  for (int rt = 0; rt < 2; ++rt)
    #pragma unroll
    for (int ct = 0; ct < 2; ++ct)
      #pragma unroll
      for (int v = 0; v < 8; ++v) {
        int row = rowbase + wrow*32 + rt*16 + v + 8*hi;
        int col = colbase + wcol*32 + ct*16 + m;
        T[(size_t)row*(C4_*CQ_) + col] = acc[rt][ct][v];
      }
}

// ---------------------------------------------------------------------------
// K5: xg[c] = inv * sum_q T[b,k,c*16+q] * Fl[b,q,k];
//     out[b,o,k] = bg[o] + sum_c Wg[o,c] * xg[c]
// ---------------------------------------------------------------------------
__global__ __launch_bounds__(64)
void k_out(const float* __restrict__ Wg, const float* __restrict__ bg,
           const float* __restrict__ ws, float* __restrict__ out) {
  __shared__ float sGq[CQ_];
  __shared__ float sXg[C4_];
  const int tid = threadIdx.x;
  const int b = blockIdx.x / N_;
  const int k = blockIdx.x % N_;
  const float* Fl = ws + OFS_FL + (size_t)b*CQ_*N_;
  const float* T  = ws + OFS_T  + (size_t)(b*N_ + k)*(C4_*CQ_);
  if (tid < CQ_) sGq[tid] = Fl[tid*N_ + k];
  __syncthreads();
  const float inv = ws[OFS_INV + b];
  {
    float s = 0.f;
    #pragma unroll
    for (int q = 0; q < CQ_; ++q) s += T[tid*CQ_ + q] * sGq[q];
    sXg[tid] = s * inv;
  }
  __syncthreads();
  float acc = bg[tid];
  #pragma unroll 8
  for (int c = 0; c < C4_; ++c) acc += Wg[tid*C4_ + c] * sXg[c];
  out[((size_t)b*C4_ + tid)*N_ + k] = acc;
}

// ---------------------------------------------------------------------------
extern "C" void kernel_launch(void* const* d_in, const int* in_sizes, int n_in,
                              void* d_out, int out_size, void* d_ws, size_t ws_size,
                              hipStream_t stream) {
  const float* x  = (const float*)d_in[0];
  const float* Wc = (const float*)d_in[1];
  const float* bc = (const float*)d_in[2];
  const float* Wg = (const float*)d_in[3];
  const float* bg = (const float*)d_in[4];
  float* ws  = (float*)d_ws;
  float* out = (float*)d_out;

  k_fc_m    <<<B_*N_, 256, 0, stream>>>(x, Wc, bc, ws);
  k_l2      <<<B_,    256, 0, stream>>>(ws);
  k_corr_max<<<B_*N_, 256, 0, stream>>>(ws);
  k_t_gemm  <<<B_*64, 256, 0, stream>>>(ws);
  k_out     <<<B_*N_,  64, 0, stream>>>(Wg, bg, ws, out);
}